// Magnus4th_65807488909634
// MI455X (gfx1250) — compile-verified
//
#include <hip/hip_runtime.h>

// Magnus 4th-order step, B=65536 batched 16x16 f32 systems, fully fused:
//   alpha build + commutator + Omega + expm(Omega) via order-8 Horner Taylor
//   (valid: ||Omega|| ~ 3e-2 for h=0.01, A~0.1*N(0,1)) + matvec + aux copy.
// One wave32 per batch matrix; matmuls via V_WMMA_F32_16X16X4_F32 (4 k-chunks).

typedef float v2f __attribute__((ext_vector_type(2)));
typedef float v8f __attribute__((ext_vector_type(8)));

#define LDS_STRIDE 17  // 16 + 1 pad: A-fragment loads (stride-16 column reads) hit 16 banks

__device__ __forceinline__ v8f wmma16x16x4(v2f a, v2f b, v8f c) {
  // (neg_a, A, neg_b, B, c_mod, C, reuse_a, reuse_b); f32 has no A/B neg -> false
  return __builtin_amdgcn_wmma_f32_16x16x4_f32(false, a, false, b, (short)0, c,
                                               false, false);
}

__global__ __launch_bounds__(256, 2) void magnus4_kernel(
    const float* __restrict__ gA1, const float* __restrict__ gA2,
    const float* __restrict__ gy0, const float* __restrict__ gh,
    float* __restrict__ y_next, float* __restrict__ aux1,
    float* __restrict__ aux2, int nBatch) {
  __shared__ float ldsOm[8][16 * LDS_STRIDE];  // alpha1, later Omega
  __shared__ float ldsS[8][16 * LDS_STRIDE];   // alpha2, later Horner iterate T

  const int tid = threadIdx.x;
  const int wave = tid >> 5;
  const int lane = tid & 31;
  const int half = lane >> 4;   // 0: lanes 0-15, 1: lanes 16-31
  const int lcol = lane & 15;   // N index in B/C/D layout, M index in A layout
  const int batch = blockIdx.x * 8 + wave;
  if (batch >= nBatch) return;  // wave-uniform; never taken for B=65536

  float* Om = ldsOm[wave];
  float* Sm = ldsS[wave];

  const float h = gh[0];
  const float c1 = 0.5f * h;
  const float c2 = 1.7320508075688772f * h;  // h*sqrt(3)

  // ---- Fused load: A1/A2 in (b128), aux out (b128), alphas -> LDS ----
  {
    const size_t mb = (size_t)batch * 256;
    const float4* a1v = reinterpret_cast<const float4*>(gA1 + mb);
    const float4* a2v = reinterpret_cast<const float4*>(gA2 + mb);
    float4* o1v = reinterpret_cast<float4*>(aux1 + mb);
    float4* o2v = reinterpret_cast<float4*>(aux2 + mb);
#pragma unroll
    for (int i = 0; i < 2; ++i) {
      const int v = lane * 2 + i;  // float4 index; flat element base = 4*v
      float4 x = a1v[v];
      float4 y = a2v[v];
      o1v[v] = x;  // aux = stack(A1, A2)
      o2v[v] = y;
      const int flat = v * 4;
      const int row = flat >> 4;
      const int col = flat & 15;
      float* om = Om + row * LDS_STRIDE + col;
      float* sm = Sm + row * LDS_STRIDE + col;
      om[0] = c1 * (x.x + y.x); sm[0] = c2 * (y.x - x.x);
      om[1] = c1 * (x.y + y.y); sm[1] = c2 * (y.y - x.y);
      om[2] = c1 * (x.z + y.z); sm[2] = c2 * (y.z - x.z);
      om[3] = c1 * (x.w + y.w); sm[3] = c2 * (y.w - x.w);
    }
  }
  __syncthreads();

  // A-matrix 16x4 f32 fragment: lanes 0-15 get K=k0,k0+1; lanes 16-31 K=k0+2,k0+3
  auto loadA = [&](const float* X, int kc) {
    const int k0 = 4 * kc + 2 * half;
    v2f a;
    a.x = X[lcol * LDS_STRIDE + k0];
    a.y = X[lcol * LDS_STRIDE + k0 + 1];
    return a;
  };
  // B-matrix 4x16 f32 fragment (row striped across lanes, K split lo/hi half-wave)
  auto loadB = [&](const float* X, int kc) {
    const int k0 = 4 * kc + 2 * half;
    v2f b;
    b.x = X[k0 * LDS_STRIDE + lcol];
    b.y = X[(k0 + 1) * LDS_STRIDE + lcol];
    return b;
  };

  // ---- commutator: P = a1@a2, Q = a2@a1 (4 x WMMA 16x16x4 each) ----
  v8f P = {};
  v8f Q = {};
#pragma unroll
  for (int kc = 0; kc < 4; ++kc) P = wmma16x16x4(loadA(Om, kc), loadB(Sm, kc), P);
#pragma unroll
  for (int kc = 0; kc < 4; ++kc) Q = wmma16x16x4(loadA(Sm, kc), loadB(Om, kc), Q);

  // ---- Omega = alpha1 - (P - Q)/12, in C/D layout regs; write back to LDS ----
  float omreg[8];
#pragma unroll
  for (int r = 0; r < 8; ++r) {
    const int row = r + 8 * half;
    omreg[r] = Om[row * LDS_STRIDE + lcol] - (P[r] - Q[r]) * (1.0f / 12.0f);
  }
  __syncthreads();
#pragma unroll
  for (int r = 0; r < 8; ++r)
    Om[(r + 8 * half) * LDS_STRIDE + lcol] = omreg[r];
  __syncthreads();

  // Loop-invariant A-fragments of Omega stay in VGPRs for the whole Horner loop
  v2f af[4];
#pragma unroll
  for (int kc = 0; kc < 4; ++kc) af[kc] = loadA(Om, kc);

  // ---- U = expm(Omega) via Horner Taylor order 8: T = I + (Omega@T)/k ----
  float Treg[8];
#pragma unroll
  for (int r = 0; r < 8; ++r)
    Treg[r] = (lcol == r + 8 * half) ? 1.0f : 0.0f;

#pragma unroll
  for (int k = 8; k >= 1; --k) {
#pragma unroll
    for (int r = 0; r < 8; ++r)
      Sm[(r + 8 * half) * LDS_STRIDE + lcol] = Treg[r];
    __syncthreads();
    v8f M = {};
#pragma unroll
    for (int kc = 0; kc < 4; ++kc) M = wmma16x16x4(af[kc], loadB(Sm, kc), M);
    const float inv = 1.0f / (float)k;  // constant-folded (k is unrolled)
#pragma unroll
    for (int r = 0; r < 8; ++r)
      Treg[r] = ((lcol == r + 8 * half) ? 1.0f : 0.0f) + M[r] * inv;
    __syncthreads();
  }

  // ---- y_next = U @ y0: per-lane products, shfl-xor reduce within 16-lane halves ----
  const float yv = gy0[(size_t)batch * 16 + lcol];
  float acc[8];
#pragma unroll
  for (int r = 0; r < 8; ++r) acc[r] = Treg[r] * yv;
#pragma unroll
  for (int m = 1; m < 16; m <<= 1) {
#pragma unroll
    for (int r = 0; r < 8; ++r) acc[r] += __shfl_xor(acc[r], m, 16);
  }
  if (lane == 0) {
#pragma unroll
    for (int r = 0; r < 8; ++r) y_next[(size_t)batch * 16 + r] = acc[r];
  } else if (lane == 16) {
#pragma unroll
    for (int r = 0; r < 8; ++r) y_next[(size_t)batch * 16 + 8 + r] = acc[r];
  }
}

extern "C" void kernel_launch(void* const* d_in, const int* in_sizes, int n_in,
                              void* d_out, int out_size, void* d_ws,
                              size_t ws_size, hipStream_t stream) {
  const float* A1 = (const float*)d_in[0];
  const float* A2 = (const float*)d_in[1];
  const float* y0 = (const float*)d_in[2];
  const float* h = (const float*)d_in[3];

  const int nBatch = in_sizes[0] / 256;  // 65536
  float* out = (float*)d_out;
  float* y_next = out;                                  // nBatch*16
  float* aux1 = out + (size_t)nBatch * 16;              // nBatch*256 (A1 copy)
  float* aux2 = aux1 + (size_t)nBatch * 256;            // nBatch*256 (A2 copy)

  const int blocks = (nBatch + 7) / 8;  // 8 waves (= 8 batches) per 256-thread block
  magnus4_kernel<<<blocks, 256, 0, stream>>>(A1, A2, y0, h, y_next, aux1, aux2,
                                             nBatch);
}